// GraphGATConv_45157286150945
// MI455X (gfx1250) — compile-verified
//
#include <hip/hip_runtime.h>
#include <hip/hip_bf16.h>

#define N_NODES 50000
#define E_EDGES 800000
#define HID     128
#define HEADS   4
#define DHEAD   32
#define NEG_SLOPE 0.2f
#define LN_EPS    1e-5f

typedef __attribute__((ext_vector_type(2))) float v2f;
typedef __attribute__((ext_vector_type(8))) float v8f;

// Non-returning, agent-scope fp32 atomic add -> native global_atomic_add_f32.
__device__ __forceinline__ void atomic_add_f32(float* addr, float val) {
  __hip_atomic_fetch_add(addr, val, __ATOMIC_RELAXED, __HIP_MEMORY_SCOPE_AGENT);
}

// ---------------------------------------------------------------------------
// GEMM: H = X @ W  (X:[N,128] row-major, W:[128,128] row-major, H:[N,128])
// One wave computes a 16x16 output tile with V_WMMA_F32_16X16X4_F32.
// Block = 256 threads = 8 waves = one 16-row strip across all 8 col tiles.
// N = 50000 = 3125 * 16 exactly -> no bounds checks.
// ---------------------------------------------------------------------------
__global__ __launch_bounds__(256) void gemm_wmma_kernel(
    const float* __restrict__ X, const float* __restrict__ W,
    float* __restrict__ H) {
  const int lane = threadIdx.x & 31;
  const int wave = threadIdx.x >> 5;        // col tile 0..7
  const int rowBase = blockIdx.x * 16;
  const int colBase = wave * 16;
  const int half = lane >> 4;               // 0: lanes 0-15, 1: lanes 16-31
  const int l = lane & 15;

  v8f c = {};
  const float* xrow = X + (size_t)(rowBase + l) * HID;
  const float* wcol = W + colBase + l;

  #pragma unroll 8
  for (int k = 0; k < HID; k += 4) {
    const int ka = k + half * 2;            // A: lanes 0-15 -> K=k,k+1 ; 16-31 -> K=k+2,k+3
    v2f a = *(const v2f*)(xrow + ka);       // 8B aligned (row base 512B aligned, ka even)
    v2f b;
    b.x = wcol[(size_t)(ka)     * HID];     // B: row-of-W striped across lanes
    b.y = wcol[(size_t)(ka + 1) * HID];
    c = __builtin_amdgcn_wmma_f32_16x16x4_f32(
        /*neg_a=*/false, a, /*neg_b=*/false, b,
        /*c_mod=*/(short)0, c, /*reuse_a=*/false, /*reuse_b=*/false);
  }

  // C layout: VGPR i -> lanes 0-15: M=i, lanes 16-31: M=8+i ; N = lane&15
  float* hb = H + (size_t)(rowBase + half * 8) * HID + colBase + l;
  #pragma unroll
  for (int i = 0; i < 8; ++i) hb[(size_t)i * HID] = c[i];
}

// ---------------------------------------------------------------------------
// el/er: per (node, head) dot of h-slice with attention vectors
// ---------------------------------------------------------------------------
__global__ __launch_bounds__(256) void eler_kernel(
    const float* __restrict__ H, const float* __restrict__ al,
    const float* __restrict__ ar, float* __restrict__ el,
    float* __restrict__ er) {
  const int t = blockIdx.x * blockDim.x + threadIdx.x;
  if (t >= N_NODES * HEADS) return;
  const int n = t >> 2, hd = t & 3;
  const float* hp  = H + (size_t)n * HID + hd * DHEAD;
  const float* alp = al + hd * DHEAD;
  const float* arp = ar + hd * DHEAD;
  float sl = 0.f, sr = 0.f;
  #pragma unroll
  for (int d = 0; d < DHEAD; d += 4) {
    float4 v = *(const float4*)(hp + d);
    float4 a = *(const float4*)(alp + d);
    float4 r = *(const float4*)(arp + d);
    sl += v.x * a.x + v.y * a.y + v.z * a.z + v.w * a.w;
    sr += v.x * r.x + v.y * r.y + v.z * r.z + v.w * r.w;
  }
  el[t] = sl;
  er[t] = sr;
}

// ---------------------------------------------------------------------------
// init: acc = 0, m = -inf, denom = 0
// ---------------------------------------------------------------------------
__global__ __launch_bounds__(256) void init_kernel(
    float* __restrict__ acc, float* __restrict__ m, float* __restrict__ denom) {
  const int t = blockIdx.x * blockDim.x + threadIdx.x;
  const int nAcc = N_NODES * HID;
  const int nH   = N_NODES * HEADS;
  if (t < nAcc)                acc[t] = 0.f;
  else if (t < nAcc + nH)      m[t - nAcc] = __int_as_float(0xff800000); // -inf
  else if (t < nAcc + 2 * nH)  denom[t - nAcc - nH] = 0.f;
}

// ---------------------------------------------------------------------------
// Edge pass 1: e = leaky_relu(el[src]+er[dst]); atomic float-max into m[dst]
// (signed/unsigned int atomic trick -> native global_atomic_{max_i32,min_u32})
// ---------------------------------------------------------------------------
__global__ __launch_bounds__(256) void edge_att_kernel(
    const int* __restrict__ src, const int* __restrict__ dst,
    const float* __restrict__ el, const float* __restrict__ er,
    float* __restrict__ ebuf, float* __restrict__ m) {
  const int t = blockIdx.x * blockDim.x + threadIdx.x;
  if (t >= E_EDGES * HEADS) return;
  const int e = t >> 2, hd = t & 3;
  const int s = src[e], d = dst[e];
  float v = el[s * HEADS + hd] + er[d * HEADS + hd];
  v = v > 0.f ? v : NEG_SLOPE * v;
  ebuf[t] = v;
  float* addr = m + d * HEADS + hd;
  if (v >= 0.f) atomicMax((int*)addr, __float_as_int(v));
  else          atomicMin((unsigned int*)addr, __float_as_uint(v));
}

// ---------------------------------------------------------------------------
// Edge pass 2: ex = exp(e - m[dst]); denom[dst] += ex  (overwrite ebuf)
// ---------------------------------------------------------------------------
__global__ __launch_bounds__(256) void edge_exp_kernel(
    const int* __restrict__ dst, float* __restrict__ ebuf,
    const float* __restrict__ m, float* __restrict__ denom) {
  const int t = blockIdx.x * blockDim.x + threadIdx.x;
  if (t >= E_EDGES * HEADS) return;
  const int e = t >> 2, hd = t & 3;
  const int d = dst[e];
  float ex = __expf(ebuf[t] - m[d * HEADS + hd]);
  ebuf[t] = ex;
  atomic_add_f32(denom + d * HEADS + hd, ex);
}

// ---------------------------------------------------------------------------
// Edge pass 3 (heavy): acc[dst, f] += h[src, f] * alpha(edge, head(f))
// One thread per (edge, feature): 128 consecutive threads share one edge ->
// coalesced gather of h[src] and coalesced atomic scatter (L2-resident).
// ---------------------------------------------------------------------------
__global__ __launch_bounds__(256) void edge_aggr_kernel(
    const int* __restrict__ src, const int* __restrict__ dst,
    const float* __restrict__ H, const float* __restrict__ ebuf,
    const float* __restrict__ denom, float* __restrict__ acc) {
  const int t = blockIdx.x * blockDim.x + threadIdx.x;  // < E*128 = 102.4M
  const int e = t >> 7, f = t & 127;
  const int hd = f >> 5;
  const int s = src[e], d = dst[e];
  const float den = denom[d * HEADS + hd];
  const float alpha = ebuf[e * HEADS + hd] / fmaxf(den, 1e-9f);
  atomic_add_f32(acc + (size_t)d * HID + f, H[(size_t)s * HID + f] * alpha);
}

// ---------------------------------------------------------------------------
// Finalize: x = acc + b; LayerNorm(g, be); ELU.  One wave32 per node,
// float4 per lane, __shfl_xor reductions across 32 lanes.
// ---------------------------------------------------------------------------
__global__ __launch_bounds__(256) void ln_elu_kernel(
    const float* __restrict__ acc, const float* __restrict__ b,
    const float* __restrict__ g, const float* __restrict__ be,
    float* __restrict__ out) {
  const int node = blockIdx.x * 8 + (threadIdx.x >> 5);
  const int lane = threadIdx.x & 31;
  if (node >= N_NODES) return;

  float4 x  = *(const float4*)(acc + (size_t)node * HID + lane * 4);
  float4 bb = *(const float4*)(b + lane * 4);
  x.x += bb.x; x.y += bb.y; x.z += bb.z; x.w += bb.w;

  float s = x.x + x.y + x.z + x.w;
  #pragma unroll
  for (int o = 16; o > 0; o >>= 1) s += __shfl_xor(s, o, 32);
  const float mu = s * (1.f / 128.f);

  float4 dx = {x.x - mu, x.y - mu, x.z - mu, x.w - mu};
  float vs = dx.x * dx.x + dx.y * dx.y + dx.z * dx.z + dx.w * dx.w;
  #pragma unroll
  for (int o = 16; o > 0; o >>= 1) vs += __shfl_xor(vs, o, 32);
  const float inv = rsqrtf(vs * (1.f / 128.f) + LN_EPS);

  float4 gg = *(const float4*)(g + lane * 4);
  float4 ee = *(const float4*)(be + lane * 4);
  float4 y;
  y.x = dx.x * inv * gg.x + ee.x;
  y.y = dx.y * inv * gg.y + ee.y;
  y.z = dx.z * inv * gg.z + ee.z;
  y.w = dx.w * inv * gg.w + ee.w;
  // ELU (alpha = 1)
  y.x = y.x > 0.f ? y.x : __expf(y.x) - 1.f;
  y.y = y.y > 0.f ? y.y : __expf(y.y) - 1.f;
  y.z = y.z > 0.f ? y.z : __expf(y.z) - 1.f;
  y.w = y.w > 0.f ? y.w : __expf(y.w) - 1.f;

  *(float4*)(out + (size_t)node * HID + lane * 4) = y;
}

// ---------------------------------------------------------------------------
// Host-side orchestration
// ---------------------------------------------------------------------------
static void run_layer(const float* X, const int* src, const int* dst,
                      const float* W, const float* al, const float* ar,
                      const float* b, const float* g, const float* be,
                      float* h, float* el, float* er, float* m, float* denom,
                      float* ebuf, float* acc, float* out, hipStream_t stream) {
  gemm_wmma_kernel<<<N_NODES / 16, 256, 0, stream>>>(X, W, h);
  eler_kernel<<<(N_NODES * HEADS + 255) / 256, 256, 0, stream>>>(h, al, ar, el, er);
  init_kernel<<<(N_NODES * (HID + 2 * HEADS) + 255) / 256, 256, 0, stream>>>(acc, m, denom);
  edge_att_kernel<<<(E_EDGES * HEADS + 255) / 256, 256, 0, stream>>>(src, dst, el, er, ebuf, m);
  edge_exp_kernel<<<(E_EDGES * HEADS + 255) / 256, 256, 0, stream>>>(dst, ebuf, m, denom);
  edge_aggr_kernel<<<(E_EDGES * HID) / 256, 256, 0, stream>>>(src, dst, h, ebuf, denom, acc);
  ln_elu_kernel<<<(N_NODES + 7) / 8, 256, 0, stream>>>(acc, b, g, be, out);
}

extern "C" void kernel_launch(void* const* d_in, const int* in_sizes, int n_in,
                              void* d_out, int out_size, void* d_ws, size_t ws_size,
                              hipStream_t stream) {
  const float* features = (const float*)d_in[0];
  const int*   src      = (const int*)d_in[1];
  const int*   dst      = (const int*)d_in[2];
  const float* W0  = (const float*)d_in[3];
  const float* al0 = (const float*)d_in[4];
  const float* ar0 = (const float*)d_in[5];
  const float* b0  = (const float*)d_in[6];
  const float* g0  = (const float*)d_in[7];
  const float* be0 = (const float*)d_in[8];
  const float* W1  = (const float*)d_in[9];
  const float* al1 = (const float*)d_in[10];
  const float* ar1 = (const float*)d_in[11];
  const float* b1  = (const float*)d_in[12];
  const float* g1  = (const float*)d_in[13];
  const float* be1 = (const float*)d_in[14];

  float* ws = (float*)d_ws;
  size_t off = 0;
  float* h     = ws + off; off += (size_t)N_NODES * HID;     // 6.4M
  float* el    = ws + off; off += (size_t)N_NODES * HEADS;   // 200K
  float* er    = ws + off; off += (size_t)N_NODES * HEADS;
  float* m     = ws + off; off += (size_t)N_NODES * HEADS;
  float* denom = ws + off; off += (size_t)N_NODES * HEADS;
  float* ebuf  = ws + off; off += (size_t)E_EDGES * HEADS;   // 3.2M
  float* acc   = ws + off; off += (size_t)N_NODES * HID;     // 6.4M
  float* x1    = ws + off; off += (size_t)N_NODES * HID;     // 6.4M

  // Layer 0: features -> x1
  run_layer(features, src, dst, W0, al0, ar0, b0, g0, be0,
            h, el, er, m, denom, ebuf, acc, x1, stream);
  // Layer 1: x1 -> d_out
  run_layer(x1, src, dst, W1, al1, ar1, b1, g1, be1,
            h, el, er, m, denom, ebuf, acc, (float*)d_out, stream);
}